// ECELoss_27419071218058
// MI455X (gfx1250) — compile-verified
//
#include <hip/hip_runtime.h>
#include <stdint.h>

typedef __attribute__((ext_vector_type(2))) float v2f;
typedef __attribute__((ext_vector_type(8))) float v8f;

#define NBINS 15
#define KC 64                  // classes (floats) staged per tile
#define ROWS_PER_WAVE 16
#define WAVES_PER_BLOCK 8
#define LDS_ROW_STRIDE 272     // 64 floats + 4 pad floats (bank-conflict-free reads)
#define BUF_BYTES (ROWS_PER_WAVE * LDS_ROW_STRIDE)   // 4352 B per buffer

// ---- CDNA5 async global->LDS (ASYNCcnt path) ----
__device__ __forceinline__ void async_ld_b128(unsigned lds_addr, const void* gaddr) {
  asm volatile("global_load_async_to_lds_b128 %0, %1, off"
               :: "v"(lds_addr), "v"(gaddr) : "memory");
}

// Stage one 16-row x 256B tile with 8 async b128 instructions.
// Lane l covers row (l>>4) parity, 16B chunk (l&15); row pointer walks by 2*C floats.
__device__ __forceinline__ void stage_tile(const float* g, unsigned lds_addr,
                                           size_t stepFloats, bool active) {
  if (active) {
#pragma unroll
    for (int j = 0; j < 8; ++j) {
      async_ld_b128(lds_addr, g);
      g += stepFloats;                 // advance 2 rows
      lds_addr += 2 * LDS_ROW_STRIDE;
    }
  }
}

// One 64-class tile of the fused exp-stream + WMMA row-sum + branchless max/argmax.
// GUARDED: clamp out-of-range classes to -inf (exp -> 0, never wins the max).
template <bool GUARDED>
__device__ __forceinline__ void compute_tile(const unsigned char* __restrict__ ldsRow,
                                             int kbase, int C, int h,
                                             float& m, int& mi, v8f& acc,
                                             const v2f& ones) {
#pragma unroll
  for (int t2 = 0; t2 < 16; ++t2) {
    const float2 v = *(const float2*)(ldsRow + 16 * t2);
    const int c0 = kbase + 4 * t2 + 2 * h;
    float x0 = v.x, x1 = v.y;
    if (GUARDED) {
      x0 = (c0 < C) ? x0 : -__builtin_inff();
      x1 = (c0 + 1 < C) ? x1 : -__builtin_inff();
    }
    const float e0 = __expf(x0);
    const float e1 = __expf(x1);
    // Reuse the compare for both index and value update (1 cmp + 2 cndmask per elem).
    const bool g0 = x0 > m;  mi = g0 ? c0 : mi;      m = g0 ? x0 : m;
    const bool g1 = x1 > m;  mi = g1 ? c0 + 1 : mi;  m = g1 ? x1 : m;
    v2f a = {e0, e1};
    acc = __builtin_amdgcn_wmma_f32_16x16x4_f32(false, a, false, ones,
                                                (short)0, acc, false, false);
  }
}

__global__ void ece_ws_init(float* ws) {
  int i = blockIdx.x * blockDim.x + threadIdx.x;
  if (i < 3 * NBINS) ws[i] = 0.0f;
}

// Pass 1: one wave owns 16 rows; double-buffered async staging; exact f32 row sums
// of exp() on the matrix pipe via V_WMMA_F32_16X16X4_F32 with B = ones.
__global__ __launch_bounds__(256) void ece_pass1(const float* __restrict__ logits,
                                                 const int* __restrict__ labels,
                                                 float* __restrict__ ws,
                                                 int N, int C) {
  __shared__ __align__(16) unsigned char stage[WAVES_PER_BLOCK * 2 * BUF_BYTES];
  __shared__ float bins[3 * NBINS];

  const int tid = threadIdx.x;
  if (tid < 3 * NBINS) bins[tid] = 0.0f;
  __syncthreads();

  const int lane = tid & 31;
  const int wave = tid >> 5;
  const int h = lane >> 4;         // K-half this lane feeds (WMMA A layout)
  const int q = lane & 15;         // row this lane scans for max/argmax
  const int schunk = lane & 15;    // staging: 16B chunk within a row
  const int srow_hi = lane >> 4;   // staging: row parity

  const unsigned sbase = (unsigned)(uintptr_t)(void*)stage;   // LDS byte offset
  const unsigned wbase = sbase + (unsigned)(wave * 2 * BUF_BYTES);
  const unsigned ldsStageOff = (unsigned)(srow_hi * LDS_ROW_STRIDE + schunk * 16);
  const unsigned char* const bufPtr0 = stage + wave * 2 * BUF_BYTES;
  const int rowReadOff = q * LDS_ROW_STRIDE + 8 * h;

  const int rowGroups = N / ROWS_PER_WAVE;
  const int gwave = blockIdx.x * WAVES_PER_BLOCK + wave;
  const int gstride = gridDim.x * WAVES_PER_BLOCK;
  const int nTiles = (C + KC - 1) / KC;
  const int tailFloats = C - (nTiles - 1) * KC;       // 1..64
  const int tailChunks = (tailFloats + 3) >> 2;
  const size_t stepFloats = 2 * (size_t)C;            // 2 rows per async step

  const v2f ones = {1.0f, 1.0f};

  for (int rg = gwave; rg < rowGroups; rg += gstride) {
    const size_t rowBase = (size_t)rg * ROWS_PER_WAVE;
    const float* const gRow =
        logits + (rowBase + (size_t)srow_hi) * (size_t)C + schunk * 4;

    // Prologue: stage tile 0 into buffer 0.
    stage_tile(gRow, wbase + ldsStageOff, stepFloats,
               schunk < (nTiles > 1 ? 16 : tailChunks));

    float m = -3.402823466e+38f;
    int mi = 0;
    v8f acc = {0.f, 0.f, 0.f, 0.f, 0.f, 0.f, 0.f, 0.f};

    // Steady state: stage t+1, wait for t, compute full tile t (no guards).
    for (int t = 0; t < nTiles - 1; ++t) {
      const int nCh = (t + 1 == nTiles - 1) ? tailChunks : 16;
      stage_tile(gRow + (t + 1) * KC,
                 wbase + (unsigned)(((t + 1) & 1) * BUF_BYTES) + ldsStageOff,
                 stepFloats, schunk < nCh);
      asm volatile("s_wait_asynccnt 8" ::: "memory");   // tile t has landed
      compute_tile<false>(bufPtr0 + (t & 1) * BUF_BYTES + rowReadOff,
                          t * KC, C, h, m, mi, acc, ones);
    }
    asm volatile("s_wait_asynccnt 0" ::: "memory");
    compute_tile<true>(bufPtr0 + ((nTiles - 1) & 1) * BUF_BYTES + rowReadOff,
                       (nTiles - 1) * KC, C, h, m, mi, acc, ones);

    // Combine max/argmax across the two K-halves (lanes q and q+16 share row q).
    {
      const float om = __shfl_xor(m, 16, 32);
      const int oi = __shfl_xor(mi, 16, 32);
      if (om > m || (om == m && oi < mi)) { m = om; mi = oi; }
    }

    // Row sums from D: lanes 0-15 hold rows 0-7 in acc[0..7], lanes 16-31 rows 8-15.
    const float a0 = acc[0], a1 = acc[1], a2 = acc[2], a3 = acc[3];
    const float a4 = acc[4], a5 = acc[5], a6 = acc[6], a7 = acc[7];
    const int q8 = lane & 7;
    float sSel = a0;
    sSel = (q8 == 1) ? a1 : sSel;
    sSel = (q8 == 2) ? a2 : sSel;
    sSel = (q8 == 3) ? a3 : sSel;
    sSel = (q8 == 4) ? a4 : sSel;
    sSel = (q8 == 5) ? a5 : sSel;
    sSel = (q8 == 6) ? a6 : sSel;
    sSel = (q8 == 7) ? a7 : sSel;
    const float sOth = __shfl_xor(sSel, 16, 32);

    if (lane < 16) {                      // lane L finalizes row L of this group
      const float s = (lane < 8) ? sSel : sOth;
      const float conf = __expf(m) / s;   // softmax max prob (unshifted exp: fp32-safe)
      const int lbl = labels[rowBase + (size_t)lane];
      const float hit = (mi == lbl) ? 1.0f : 0.0f;
      int b = (int)ceilf(conf * (float)NBINS) - 1;
      b = b < 0 ? 0 : (b > NBINS - 1 ? NBINS - 1 : b);
      atomicAdd(&bins[b], 1.0f);
      atomicAdd(&bins[NBINS + b], conf);
      atomicAdd(&bins[2 * NBINS + b], hit);
    }
  }

  __syncthreads();
  if (tid < 3 * NBINS) atomicAdd(&ws[tid], bins[tid]);
}

// Pass 2: fold 45 accumulators into ece + per-bin diagnostics (31 outputs).
__global__ void ece_finalize(const float* __restrict__ ws, float* __restrict__ out, int N) {
  if (threadIdx.x == 0 && blockIdx.x == 0) {
    float ece = 0.0f;
    const float invN = 1.0f / (float)N;
    for (int i = 0; i < NBINS; ++i) {
      const float c = ws[i];
      const float sc = ws[NBINS + i];
      const float sa = ws[2 * NBINS + i];
      const bool nonempty = c > 0.0f;
      const float safe = nonempty ? c : 1.0f;
      const float avgc = sc / safe;
      const float avga = sa / safe;
      if (nonempty) ece += fabsf(avgc - avga) * (c * invN);
      out[1 + i] = nonempty ? avgc : (((float)i + 0.5f) / (float)NBINS);
      out[16 + i] = nonempty ? avga : 0.0f;
    }
    out[0] = ece;
  }
}

extern "C" void kernel_launch(void* const* d_in, const int* in_sizes, int n_in,
                              void* d_out, int out_size, void* d_ws, size_t ws_size,
                              hipStream_t stream) {
  const float* logits = (const float*)d_in[0];
  const int* labels = (const int*)d_in[1];
  float* ws = (float*)d_ws;
  float* out = (float*)d_out;

  const int N = in_sizes[1];        // 262144
  const int C = in_sizes[0] / N;    // 1000

  ece_ws_init<<<1, 64, 0, stream>>>(ws);

  int rowGroups = N / ROWS_PER_WAVE;
  int blocks = (rowGroups + WAVES_PER_BLOCK - 1) / WAVES_PER_BLOCK;
  if (blocks > 1024) blocks = 1024;
  ece_pass1<<<blocks, 256, 0, stream>>>(logits, labels, ws, N, C);

  ece_finalize<<<1, 32, 0, stream>>>(ws, out, N);
}